// SpectralConv3d_83253646065782
// MI455X (gfx1250) — compile-verified
//
#include <hip/hip_runtime.h>

// ---------------------------------------------------------------------------
// Spectral conv via separable Hartley (cas) transforms on gfx1250 (MI455X).
// Matmuls: V_WMMA_F32_16X16X4_F32 (fp32 WMMA, wave32).
// Data movement: async LDS DMA (GLOBAL_{LOAD,STORE}_ASYNC_*_LDS, ASYNCcnt)
// when the toolchain exposes the builtins; regular load/store fallback else.
// ---------------------------------------------------------------------------

typedef __attribute__((ext_vector_type(2))) float v2f;
typedef __attribute__((ext_vector_type(8))) float v8f;
typedef __attribute__((ext_vector_type(4))) int v4i;

#define NFREQ 110592  // 48^3

#if defined(__has_builtin)
#if __has_builtin(__builtin_amdgcn_global_load_async_to_lds_b32) &&     \
    __has_builtin(__builtin_amdgcn_global_load_async_to_lds_b128) &&    \
    __has_builtin(__builtin_amdgcn_global_store_async_from_lds_b128)
#define USE_ASYNC 1
#endif
#endif
#ifndef USE_ASYNC
#define USE_ASYNC 0
#endif

#if USE_ASYNC
// Per the compiler diagnostics: b32 takes global 'int*' (AS1, prints as
// '__device__ int *'), b128 takes global 'v4i*'; LDS side is AS3.
typedef __attribute__((address_space(1))) int* g_i32p;
typedef __attribute__((address_space(1))) v4i* g_v4ip;
typedef __attribute__((address_space(3))) int* l_i32p;
typedef __attribute__((address_space(3))) v4i* l_v4ip;

__device__ __forceinline__ void async_ld_b32(const void* g, void* l) {
  __builtin_amdgcn_global_load_async_to_lds_b32((g_i32p)g, (l_i32p)l, 0, 0);
}
__device__ __forceinline__ void async_ld_b128(const void* g, void* l) {
  __builtin_amdgcn_global_load_async_to_lds_b128((g_v4ip)g, (l_v4ip)l, 0, 0);
}
__device__ __forceinline__ void async_st_b128(void* g, void* l) {
  __builtin_amdgcn_global_store_async_from_lds_b128((g_v4ip)g, (l_v4ip)l, 0, 0);
}
#endif

__device__ __forceinline__ void wait_async0() {
#if defined(__has_builtin) && __has_builtin(__builtin_amdgcn_s_wait_asynccnt)
  __builtin_amdgcn_s_wait_asynccnt(0);
#else
  asm volatile("s_wait_asynccnt 0x0" ::: "memory");
#endif
}

__device__ __forceinline__ v8f wmma_f32_4(v2f a, v2f b, v8f c) {
  // 8 args: (neg_a, A, neg_b, B, c_mod, C, reuse_a, reuse_b)
  return __builtin_amdgcn_wmma_f32_16x16x4_f32(false, a, false, b, (short)0, c,
                                               false, false);
}

// ---------------------------------------------------------------------------
// One separable transform pass.  Per volume v:
//   in  shape (A, B, KIN)  row-major, lines along last axis (length KIN)
//   out shape (48, A, B):  out[k][a][b] = sum_j in[a][b][j] * cas(2*pi*j*k/48)
// R = A*B rows; each block handles 128 rows of one volume.
// KIN=48 for data passes, KIN=12 for the zero-padded weight block.
// inL row stride 52 floats (=13*16B) keeps async B128 dsts 16B-aligned and is
// bank-conflict free for A-fragment reads (gcd(52,64)=4 -> 16 distinct banks).
// ---------------------------------------------------------------------------
template <int KIN>
__global__ __launch_bounds__(256) void dht_pass(const float* __restrict__ in,
                                                float* __restrict__ out,
                                                int R, long strideIn,
                                                long strideOut, float scale) {
  __shared__ float casL[48 * 49];   // cas[j][k]
  __shared__ float inL[128 * 52];   // [localRow][j], 16B-aligned rows
  __shared__ float outL[48 * 132];  // [k][localRow], 132 = 33*4 -> 16B rows

  const int tid = threadIdx.x;
  const int v = blockIdx.y;
  const int rBase = blockIdx.x * 128;
  const float* vin = in + (long)v * strideIn;
  float* vout = out + (long)v * strideOut;

  // Stage 128 input lines into LDS.
#if USE_ASYNC
  {
    const int CH = KIN / 4;  // 16-byte chunks per line
    for (int idx = tid; idx < 128 * CH; idx += 256) {
      int lr = idx / CH, c = idx - lr * CH;
      int r = rBase + lr;
      if (r < R) {
        async_ld_b128(vin + (long)r * KIN + c * 4, &inL[lr * 52 + c * 4]);
      } else {  // zero-pad tail rows
        inL[lr * 52 + c * 4 + 0] = 0.0f;
        inL[lr * 52 + c * 4 + 1] = 0.0f;
        inL[lr * 52 + c * 4 + 2] = 0.0f;
        inL[lr * 52 + c * 4 + 3] = 0.0f;
      }
    }
  }
#else
  __builtin_prefetch(vin + (long)rBase * KIN, 0, 0);  // global_prefetch_b8
  for (int idx = tid; idx < 128 * KIN; idx += 256) {
    int lr = idx / KIN, j = idx - lr * KIN;
    int r = rBase + lr;
    inL[lr * 52 + j] = (r < R) ? vin[(long)r * KIN + j] : 0.0f;
  }
#endif

  // Build cas matrix in LDS (overlaps with async DMA).
  for (int idx = tid; idx < 48 * 48; idx += 256) {
    int j = idx / 48, k = idx - j * 48;
    float th = 6.2831853071795864f * (float)((j * k) % 48) * (1.0f / 48.0f);
    float s, c;
    __sincosf(th, &s, &c);
    casL[j * 49 + k] = c + s;
  }
#if USE_ASYNC
  wait_async0();
#endif
  __syncthreads();

  const int wv = tid >> 5;           // 8 waves, each owns 16 rows
  const int lane = tid & 31;
  const int rlo = lane & 15;         // M row within tile / N col of B
  const int hi2 = (lane >> 4) << 1;  // lanes 16-31 hold j0+2, j0+3
  const int rowBaseL = wv * 16;

  for (int kt = 0; kt < 3; ++kt) {  // 48 output freqs = 3 tiles of N=16
    v8f acc = {};
#pragma unroll
    for (int j0 = 0; j0 < KIN; j0 += 4) {
      int jj = j0 + hi2;
      v2f a, b;
      a.x = inL[(rowBaseL + rlo) * 52 + jj];      // A[r][jj]
      a.y = inL[(rowBaseL + rlo) * 52 + jj + 1];  // A[r][jj+1]
      b.x = casL[jj * 49 + kt * 16 + rlo];        // B[jj][k]
      b.y = casL[(jj + 1) * 49 + kt * 16 + rlo];  // B[jj+1][k]
      acc = wmma_f32_4(a, b, acc);
    }
    // D layout: vgpr m -> row M=m (lanes 0-15) / M=m+8 (lanes 16-31), N=rlo.
    int mOff = (lane < 16) ? 0 : 8;
#pragma unroll
    for (int m = 0; m < 8; ++m)
      outL[(kt * 16 + rlo) * 132 + rowBaseL + m + mOff] = acc[m] * scale;
  }
  __syncthreads();

  // Store: for each k, 128 consecutive floats at out[k*R + rBase].
#if USE_ASYNC
  for (int idx = tid; idx < 48 * 32; idx += 256) {
    int k = idx >> 5, c = idx & 31;  // c: 16-byte chunk within 128 rows
    int rr = rBase + c * 4;
    if (rr < R)  // R, rBase multiples of 16 -> chunks never straddle the edge
      async_st_b128(vout + (long)k * R + rr, &outL[k * 132 + c * 4]);
  }
  // s_endpgm performs an implicit wait-idle; kernel-boundary ordering is safe.
#else
  for (int idx = tid; idx < 48 * 128; idx += 256) {
    int k = idx >> 7, lr = idx & 127;
    int r = rBase + lr;
    if (r < R) vout[(long)k * R + r] = outL[k * 132 + lr];
  }
#endif
}

// ---------------------------------------------------------------------------
// Spectral channel mixing.  For each frequency f (flat over 48^3), with the
// flat periodic flip f' = (n - f) mod n:
//   Z[b,o,f] = 0.5 * sum_i (X[b,i,f]+X[b,i,f'])*Y[i,o,f]
//                   + (X[b,i,f]-X[b,i,f'])*Y[i,o,f']
// One block = 32 consecutive frequencies; each wave does 4 frequencies,
// each as two 16x16x16 fp32 WMMA GEMMs (b on M rows 0-7, i on K, o on N).
// ---------------------------------------------------------------------------
__global__ __launch_bounds__(256) void mix_kernel(const float* __restrict__ X,
                                                  const float* __restrict__ Y,
                                                  float* __restrict__ Z) {
  extern __shared__ float sm[];
  float* Xl = sm;              // [128][33]  X at f
  float* Xfl = Xl + 128 * 33;  // [128][33]  X at f'
  float* Yl = Xfl + 128 * 33;  // [256][33]  Y at f
  float* Yfl = Yl + 256 * 33;  // [256][33]  Y at f'
  float* Zl = Yfl + 256 * 33;  // [128][36]  Z staging (36=9*16B rows)

  const int tid = threadIdx.x;
  const int f0 = blockIdx.x * 32;
  const int n = NFREQ;

  for (int idx = tid; idx < 128 * 32; idx += 256) {
    int ch = idx >> 5, t = idx & 31;
    int f = f0 + t;
    int fp = (n - f) % n;  // flat flip; per-element gather
#if USE_ASYNC
    async_ld_b32(&X[(long)ch * n + f], &Xl[ch * 33 + t]);
    async_ld_b32(&X[(long)ch * n + fp], &Xfl[ch * 33 + t]);
#else
    Xl[ch * 33 + t] = X[(long)ch * n + f];
    Xfl[ch * 33 + t] = X[(long)ch * n + fp];
#endif
  }
  for (int idx = tid; idx < 256 * 32; idx += 256) {
    int ch = idx >> 5, t = idx & 31;
    int f = f0 + t;
    int fp = (n - f) % n;
#if USE_ASYNC
    async_ld_b32(&Y[(long)ch * n + f], &Yl[ch * 33 + t]);
    async_ld_b32(&Y[(long)ch * n + fp], &Yfl[ch * 33 + t]);
#else
    Yl[ch * 33 + t] = Y[(long)ch * n + f];
    Yfl[ch * 33 + t] = Y[(long)ch * n + fp];
#endif
  }
#if USE_ASYNC
  wait_async0();
#endif
  __syncthreads();

  const int wv = tid >> 5;
  const int lane = tid & 31;
  const int rlo = lane & 15;  // M row (= batch b, rows 8-15 padding)
  const int hi2 = (lane >> 4) << 1;
  const bool rvalid = rlo < 8;
  const int rr = rvalid ? rlo : 0;  // keep LDS indices in-range for pad rows

  for (int q = 0; q < 4; ++q) {
    const int t = wv * 4 + q;
    v8f acc = {};
    // term 1: (X + Xflip) * Y[f]
#pragma unroll
    for (int j0 = 0; j0 < 16; j0 += 4) {
      int jj = j0 + hi2;
      v2f a, b;
      float x0 = Xl[(rr * 16 + jj) * 33 + t], g0 = Xfl[(rr * 16 + jj) * 33 + t];
      float x1 = Xl[(rr * 16 + jj + 1) * 33 + t],
            g1 = Xfl[(rr * 16 + jj + 1) * 33 + t];
      a.x = rvalid ? (x0 + g0) : 0.0f;
      a.y = rvalid ? (x1 + g1) : 0.0f;
      b.x = Yl[(jj * 16 + rlo) * 33 + t];  // Y[i=jj][o=rlo]
      b.y = Yl[((jj + 1) * 16 + rlo) * 33 + t];
      acc = wmma_f32_4(a, b, acc);
    }
    // term 2: (X - Xflip) * Y[f']
#pragma unroll
    for (int j0 = 0; j0 < 16; j0 += 4) {
      int jj = j0 + hi2;
      v2f a, b;
      float x0 = Xl[(rr * 16 + jj) * 33 + t], g0 = Xfl[(rr * 16 + jj) * 33 + t];
      float x1 = Xl[(rr * 16 + jj + 1) * 33 + t],
            g1 = Xfl[(rr * 16 + jj + 1) * 33 + t];
      a.x = rvalid ? (x0 - g0) : 0.0f;
      a.y = rvalid ? (x1 - g1) : 0.0f;
      b.x = Yfl[(jj * 16 + rlo) * 33 + t];
      b.y = Yfl[((jj + 1) * 16 + rlo) * 33 + t];
      acc = wmma_f32_4(a, b, acc);
    }
    if (lane < 16) {  // rows M=0..7 are valid batches (vgpr m -> b=m)
#pragma unroll
      for (int m = 0; m < 8; ++m)
        Zl[(m * 16 + rlo) * 36 + t] = 0.5f * acc[m];
    }
  }
  __syncthreads();

#if USE_ASYNC
  for (int idx = tid; idx < 128 * 8; idx += 256) {
    int ch = idx >> 3, c = idx & 7;  // 8 x 16-byte chunks of the 32 freqs
    async_st_b128(&Z[(long)ch * n + f0 + c * 4], &Zl[ch * 36 + c * 4]);
  }
#else
  for (int idx = tid; idx < 128 * 32; idx += 256) {
    int ch = idx >> 5, t = idx & 31;
    Z[(long)ch * n + f0 + t] = Zl[ch * 36 + t];
  }
#endif
}

// ---------------------------------------------------------------------------
// Launcher.  Inputs: d_in[0]=x (8,16,48,48,48) f32, d_in[1]=error (unused,
// zeros -> full modes), d_in[2]=weights (16,16,12,12,12) f32.
// Workspace (floats): bufA,bufB = 128*n each; y1 = 256*6912; y2 = 256*27648;
// yb = 256*n.  Total ~262 MB.
// ---------------------------------------------------------------------------
extern "C" void kernel_launch(void* const* d_in, const int* in_sizes, int n_in,
                              void* d_out, int out_size, void* d_ws,
                              size_t ws_size, hipStream_t stream) {
  const float* x = (const float*)d_in[0];
  const float* w = (const float*)d_in[2];
  float* out = (float*)d_out;
  float* ws = (float*)d_ws;

  const long n = NFREQ;
  float* bufA = ws;
  float* bufB = bufA + 128L * n;
  float* y1 = bufB + 128L * n;   // (k3,12,12) per (i,o): 256*6912
  float* y2 = y1 + 256L * 6912;  // (k2,k3,12): 256*27648
  float* yb = y2 + 256L * 27648; // (k1,k2,k3): 256*n

  dim3 blk(256);
  const size_t mixLds = (768L * 33 + 128L * 36) * sizeof(float);  // ~119 KB

  // Forward DHT of x: 3 axis-rotating passes, 128 volumes, R=2304 -> 18 blocks
  dht_pass<48><<<dim3(18, 128), blk, 0, stream>>>(x, bufA, 2304, n, n, 1.0f);
  dht_pass<48><<<dim3(18, 128), blk, 0, stream>>>(bufA, bufB, 2304, n, n, 1.0f);
  dht_pass<48><<<dim3(18, 128), blk, 0, stream>>>(bufB, bufA, 2304, n, n, 1.0f);

  // Forward DHT of zero-padded weights (padding folded: KIN=12), 256 volumes
  dht_pass<12><<<dim3(2, 256), blk, 0, stream>>>(w, y1, 144, 1728, 6912, 1.0f);
  dht_pass<12><<<dim3(5, 256), blk, 0, stream>>>(y1, y2, 576, 6912, 27648, 1.0f);
  dht_pass<12><<<dim3(18, 256), blk, 0, stream>>>(y2, yb, 2304, 27648, n, 1.0f);

  // Spectral channel mixing (even/odd Hartley terms via flat flip)
  mix_kernel<<<dim3((int)(n / 32)), blk, mixLds, stream>>>(bufA, yb, bufB);

  // Inverse DHT = forward DHT / n (scale folded into last pass)
  dht_pass<48><<<dim3(18, 128), blk, 0, stream>>>(bufB, bufA, 2304, n, n, 1.0f);
  dht_pass<48><<<dim3(18, 128), blk, 0, stream>>>(bufA, bufB, 2304, n, n, 1.0f);
  dht_pass<48><<<dim3(18, 128), blk, 0, stream>>>(bufB, out, 2304, n, n,
                                                  1.0f / (float)n);
}